// imaging_layer_56538949484865
// MI455X (gfx1250) — compile-verified
//
#include <hip/hip_runtime.h>
#include <hip/hip_bf16.h>

typedef float v2f __attribute__((ext_vector_type(2)));
typedef float v8f __attribute__((ext_vector_type(8)));

#define MDIM   250
#define SQ     12
#define NREG   10
#define BATCH  4096

// One wave (32 lanes) per block; each block reduces 16 batches' 12x12
// sum-of-squares via V_WMMA_F32_16X16X4_F32 with an all-ones B matrix,
// then writes the L2-normalized [16 x 10] output tile.
__global__ __launch_bounds__(32)
void imaging_layer_kernel(const float* __restrict__ u, float* __restrict__ out) {
    const int lane  = threadIdx.x;        // 0..31
    const int m     = lane & 15;          // batch within tile (M row of A)
    const int half  = lane >> 4;          // selects K pair {0,1} or {2,3}
    const size_t b  = (size_t)blockIdx.x * 16 + m;

    const float* base = u + b * (size_t)(MDIM * MDIM);

    // A-matrix partials: a.x = K=2*half (rows 6h..6h+2), a.y = K=2*half+1 (rows 6h+3..6h+5)
    float acc0 = 0.0f, acc1 = 0.0f;
    const int r0 = 6 * half;
    #pragma unroll
    for (int rr = 0; rr < 3; ++rr) {
        const float* rowA = base + (size_t)(r0 + rr)     * MDIM;
        const float* rowB = base + (size_t)(r0 + 3 + rr) * MDIM;
        #pragma unroll
        for (int cc = 0; cc < 6; ++cc) {            // 12 floats = 6x float2 (8B aligned)
            v2f xa = *(const v2f*)(rowA + 2 * cc);
            v2f xb = *(const v2f*)(rowB + 2 * cc);
            acc0 += xa.x * xa.x + xa.y * xa.y;
            acc1 += xb.x * xb.x + xb.y * xb.y;
        }
    }

    v2f a;  a.x = acc0; a.y = acc1;
    v2f bo; bo.x = 1.0f; bo.y = 1.0f;               // all-ones B => row reduction over K
    v8f c = {};

    // D[m][n] = sum_k A[m][k] = full 12x12 sum-of-squares for batch m (all columns equal)
    v8f d = __builtin_amdgcn_wmma_f32_16x16x4_f32(
        /*neg_a=*/false, a, /*neg_b=*/false, bo,
        /*c_mod=*/(short)0, c, /*reuse_a=*/false, /*reuse_b=*/false);

    // Lanes 0-15 hold M=0..7 in d[0..7] (col n=lane); lanes 16-31 hold M=8..15 (col n=lane-16).
    const int n     = lane & 15;                    // region/column index
    const int mbase = half * 8;
    if (n < NREG) {
        const size_t tile = (size_t)blockIdx.x * 16;
        #pragma unroll
        for (int i = 0; i < 8; ++i) {
            float s = d[i];
            // values = [s]*10 ; norm = sqrt(10*s^2) ; out = s / norm
            float o = s / sqrtf(10.0f * s * s);
            out[(tile + mbase + i) * NREG + n] = o;
        }
    }
}

extern "C" void kernel_launch(void* const* d_in, const int* in_sizes, int n_in,
                              void* d_out, int out_size, void* d_ws, size_t ws_size,
                              hipStream_t stream) {
    (void)in_sizes; (void)n_in; (void)out_size; (void)d_ws; (void)ws_size;
    const float* u = (const float*)d_in[0];
    float* out = (float*)d_out;
    // 4096 batches / 16 per wave = 256 blocks of 1 wave each
    imaging_layer_kernel<<<BATCH / 16, 32, 0, stream>>>(u, out);
}